// RecModel_11914239279970
// MI455X (gfx1250) — compile-verified
//
#include <hip/hip_runtime.h>
#include <hip/hip_bf16.h>
#include <math.h>

// ---- problem constants ----
#define HIDDEN 128
#define OUT_N  200000
#define T_STEPS 64
#define TOPK   50
#define HIST   200

// ---- ws layout (bytes) ----
#define OFF_MASK   0u                      // 200000 f32
#define OFF_ELOG   800000u                 // 200000 f32 (logits -> exp in place)
#define OFF_H      1600000u                // 128 f32
#define OFF_C      1600512u                // 128 f32
#define OFF_SCAL   1601024u                // [0]=gmax (orderable u32), [1]=gsum f32
#define OFF_STATE  1601040u                // [0]=a_prev, [1]=fb_prev (i32)
#define OFF_CAND   1601056u                // 5000 u64 candidates
#define OFF_PART   1641056u                // 625 f32 partial sums

typedef __attribute__((ext_vector_type(2))) float v2f;
typedef __attribute__((ext_vector_type(8))) float v8f;

__device__ __forceinline__ unsigned f2ord(float f) {
    unsigned u = __float_as_uint(f);
    return (u & 0x80000000u) ? ~u : (u | 0x80000000u);
}
__device__ __forceinline__ float ord2f(unsigned k) {
    unsigned u = (k & 0x80000000u) ? (k & 0x7fffffffu) : ~k;
    return __uint_as_float(u);
}
__device__ __forceinline__ float sigm(float x) { return 1.0f / (1.0f + expf(-x)); }

// ---------------------------------------------------------------------------
// K_init: mask=1, h=c=0, state=0
// ---------------------------------------------------------------------------
__global__ __launch_bounds__(256) void k_init(float* mask, float* hbuf, float* cbuf, int* state) {
    int i = blockIdx.x * 256 + threadIdx.x;
    if (i < OUT_N) mask[i] = 1.0f;
    if (blockIdx.x == 0) {
        if (threadIdx.x < HIDDEN) { hbuf[threadIdx.x] = 0.0f; cbuf[threadIdx.x] = 0.0f; }
        if (threadIdx.x == 0) { state[0] = 0; state[1] = 0; }
    }
}

// ---------------------------------------------------------------------------
// K_cell: emb = embed[a_prev]*fb (t>0), gates = Wih@emb + Whh@h via
// V_WMMA_F32_16X16X4_F32 (vector in column N=0 of B), LSTM update -> h,c.
// Also resets the per-step softmax scalars.
//
// A layout (32-bit 16x4): lanes 0-15 hold M=lane, K={k0,k0+1}; lanes 16-31
// hold M=lane-16, K={k0+2,k0+3}.  B (4x16): V0 = rows K0 (lanes 0-15) / K2
// (lanes 16-31), V1 = rows K1 / K3; only column N=0 (lane r==0 in each half)
// carries the vector.  D column 0: lane 0 -> rows m0..m0+7, lane 16 ->
// rows m0+8..m0+15.
// ---------------------------------------------------------------------------
__global__ __launch_bounds__(256) void k_cell(const float* __restrict__ embed,
                                              const float* __restrict__ W_ih,
                                              const float* __restrict__ W_hh,
                                              const float* __restrict__ b_ih,
                                              const float* __restrict__ b_hh,
                                              float* hbuf, float* cbuf,
                                              const int* state,
                                              unsigned* gmaxU, float* gsum, int t) {
    __shared__ float v[2 * HIDDEN];     // [emb ; h]
    __shared__ float gl[4 * HIDDEN];    // gates
    int tid = threadIdx.x;
    if (tid == 0) { *gmaxU = 0u; *gsum = 0.0f; }
    if (tid < HIDDEN) {
        float e = 0.0f;
        if (t > 0) {
            int ap = state[0], fb = state[1];
            e = embed[(size_t)ap * HIDDEN + tid] * (float)fb;
        }
        v[tid] = e;
        v[HIDDEN + tid] = hbuf[tid];
    }
    __syncthreads();

    int lane = tid & 31, wave = tid >> 5;
    int half = lane >> 4, r = lane & 15;
    bool col0 = (r == 0);
    for (int j = 0; j < 4; ++j) {
        int m0 = (wave * 4 + j) * 16;      // 8 waves * 4 tiles = 32 tiles = 512 rows
        int row = m0 + r;
        const float* wih = W_ih + (size_t)row * HIDDEN + half * 2;
        const float* whh = W_hh + (size_t)row * HIDDEN + half * 2;
        v8f acc = {0.f, 0.f, 0.f, 0.f, 0.f, 0.f, 0.f, 0.f};
        // ---- first half: W_ih @ emb (v[0..127]) ----
        for (int k0 = 0; k0 < HIDDEN; k0 += 4) {
            v2f a; a.x = wih[k0 + 0]; a.y = wih[k0 + 1];
            // uniform (broadcast) LDS reads, then plain select -> v_cndmask
            float b0 = v[k0 + half * 2 + 0];
            float b1 = v[k0 + half * 2 + 1];
            v2f b; b.x = col0 ? b0 : 0.0f; b.y = col0 ? b1 : 0.0f;
            acc = __builtin_amdgcn_wmma_f32_16x16x4_f32(false, a, false, b,
                                                        (short)0, acc, false, false);
        }
        // ---- second half: W_hh @ h (v[128..255]) ----
        for (int k0 = 0; k0 < HIDDEN; k0 += 4) {
            v2f a; a.x = whh[k0 + 0]; a.y = whh[k0 + 1];
            float b0 = v[HIDDEN + k0 + half * 2 + 0];
            float b1 = v[HIDDEN + k0 + half * 2 + 1];
            v2f b; b.x = col0 ? b0 : 0.0f; b.y = col0 ? b1 : 0.0f;
            acc = __builtin_amdgcn_wmma_f32_16x16x4_f32(false, a, false, b,
                                                        (short)0, acc, false, false);
        }
        if (col0) {
#pragma unroll
            for (int q = 0; q < 8; ++q) gl[m0 + half * 8 + q] = acc[q];
        }
    }
    __syncthreads();

    if (tid < HIDDEN) {
        float ig = gl[tid]              + b_ih[tid]              + b_hh[tid];
        float fg = gl[HIDDEN + tid]     + b_ih[HIDDEN + tid]     + b_hh[HIDDEN + tid];
        float gg = gl[2 * HIDDEN + tid] + b_ih[2 * HIDDEN + tid] + b_hh[2 * HIDDEN + tid];
        float og = gl[3 * HIDDEN + tid] + b_ih[3 * HIDDEN + tid] + b_hh[3 * HIDDEN + tid];
        float cn = sigm(fg) * cbuf[tid] + sigm(ig) * tanhf(gg);
        float hn = sigm(og) * tanhf(cn);
        cbuf[tid] = cn;
        hbuf[tid] = hn;
    }
}

// ---------------------------------------------------------------------------
// K_matvec: logits = W_out @ h + b_out. One wave per row (32 lanes * float4 =
// 512B = one full row), 5000 waves x 40 rows. Tracks global max via atomicMax
// on orderable bits (deterministic).
// ---------------------------------------------------------------------------
__global__ __launch_bounds__(256) void k_matvec(const float* __restrict__ W_out,
                                                const float* __restrict__ b_out,
                                                const float* __restrict__ hbuf,
                                                float* __restrict__ elog,
                                                unsigned* gmaxU) {
    __shared__ float wred[8];
    int tid = threadIdx.x;
    int lane = tid & 31, wave = tid >> 5;
    int gw = blockIdx.x * 8 + wave;            // 625*8 = 5000 waves
    const float4 hv = *(const float4*)(hbuf + lane * 4);
    float lmax = -3.402823466e+38f;
    for (int it = 0; it < 40; ++it) {
        int row = gw + it * 5000;
        const float4 w = *(const float4*)(W_out + (size_t)row * HIDDEN + lane * 4);
        float d = w.x * hv.x + w.y * hv.y + w.z * hv.z + w.w * hv.w;
#pragma unroll
        for (int m = 16; m >= 1; m >>= 1) d += __shfl_xor(d, m, 32);
        if (lane == 0) {
            float lg = d + b_out[row];
            elog[row] = lg;
            lmax = fmaxf(lmax, lg);
        }
    }
    if (lane == 0) wred[wave] = lmax;
    __syncthreads();
    if (tid == 0) {
        float m = wred[0];
#pragma unroll
        for (int w = 1; w < 8; ++w) m = fmaxf(m, wred[w]);
        atomicMax(gmaxU, f2ord(m));
    }
}

// ---------------------------------------------------------------------------
// K_expsum: e = exp(logit - gmax) in place; per-block partial sums (deterministic).
// ---------------------------------------------------------------------------
__global__ __launch_bounds__(256) void k_expsum(float* __restrict__ elog,
                                                const unsigned* gmaxU,
                                                float* __restrict__ part) {
    __shared__ float red[256];
    int tid = threadIdx.x;
    float gmax = ord2f(*gmaxU);
    float s = 0.0f;
    for (int i = blockIdx.x * 256 + tid; i < OUT_N; i += 625 * 256) {
        float e = expf(elog[i] - gmax);
        elog[i] = e;
        s += e;
    }
    red[tid] = s;
    __syncthreads();
    for (int off = 128; off > 0; off >>= 1) {
        if (tid < off) red[tid] += red[tid + off];
        __syncthreads();
    }
    if (tid == 0) part[blockIdx.x] = red[0];
}

__global__ __launch_bounds__(256) void k_sumreduce(const float* __restrict__ part,
                                                   float* gsum) {
    __shared__ float red[256];
    int tid = threadIdx.x;
    float s = 0.0f;
    for (int i = tid; i < 625; i += 256) s += part[i];
    red[tid] = s;
    __syncthreads();
    for (int off = 128; off > 0; off >>= 1) {
        if (tid < off) red[tid] += red[tid + off];
        __syncthreads();
    }
    if (tid == 0) *gsum = red[0];
}

// ---------------------------------------------------------------------------
// K_topk_local: masked = e * inv * mask -> d_out slice; per-block top-50 via
// packed u64 keys (value bits high, ~index low => max picks highest value,
// lowest index, matching jax top_k tie-breaking for non-negative values).
// 100 blocks x 2000-element chunks.
// ---------------------------------------------------------------------------
__global__ __launch_bounds__(256) void k_topk_local(const float* __restrict__ elog,
                                                    const float* __restrict__ mask,
                                                    const float* gsum,
                                                    float* __restrict__ out_masked,
                                                    unsigned long long* __restrict__ cand) {
    __shared__ unsigned long long kv[2000];
    __shared__ unsigned long long red[256];
    int tid = threadIdx.x;
    int base = blockIdx.x * 2000;
    float inv = 1.0f / (*gsum);
#pragma unroll
    for (int k = 0; k < 8; ++k) {
        int li = tid + k * 256;
        if (li < 2000) {
            int i = base + li;
            float m = elog[i] * inv * mask[i];
            out_masked[i] = m;
            kv[li] = ((unsigned long long)__float_as_uint(m) << 32) |
                     (unsigned long long)(~(unsigned)i);
        }
    }
    __syncthreads();
    for (int s = 0; s < TOPK; ++s) {
        unsigned long long best = 0ull;
#pragma unroll
        for (int k = 0; k < 8; ++k) {
            int li = tid + k * 256;
            if (li < 2000 && kv[li] > best) best = kv[li];
        }
        red[tid] = best;
        __syncthreads();
        for (int off = 128; off > 0; off >>= 1) {
            if (tid < off) { unsigned long long o = red[tid + off]; if (o > red[tid]) red[tid] = o; }
            __syncthreads();
        }
        unsigned long long kmax = red[0];
        __syncthreads();
#pragma unroll
        for (int k = 0; k < 8; ++k) {
            int li = tid + k * 256;
            if (li < 2000 && kv[li] == kmax) kv[li] = 0ull;
        }
        if (tid == 0) cand[blockIdx.x * TOPK + s] = kmax;
        __syncthreads();
    }
}

// ---------------------------------------------------------------------------
// K_final: merge 5000 candidates -> global top-50; membership test against the
// 1-indexed history; pick a_new/feedback; update mask; write scalar outputs.
// ---------------------------------------------------------------------------
__global__ __launch_bounds__(256) void k_final(const unsigned long long* __restrict__ cand,
                                               const int* __restrict__ x,
                                               float* __restrict__ mask,
                                               int* state,
                                               float* __restrict__ out, int t) {
    __shared__ unsigned long long cl[5000];
    __shared__ unsigned long long red[256];
    __shared__ int xl[HIST];
    __shared__ unsigned topi[TOPK];
    __shared__ int hit[TOPK];
    int tid = threadIdx.x;
    for (int j = tid; j < 5000; j += 256) cl[j] = cand[j];
    for (int j = tid; j < HIST; j += 256) xl[j] = x[j];
    __syncthreads();
    for (int s = 0; s < TOPK; ++s) {
        unsigned long long best = 0ull;
#pragma unroll
        for (int k = 0; k < 20; ++k) {
            int li = tid + k * 256;
            if (li < 5000 && cl[li] > best) best = cl[li];
        }
        red[tid] = best;
        __syncthreads();
        for (int off = 128; off > 0; off >>= 1) {
            if (tid < off) { unsigned long long o = red[tid + off]; if (o > red[tid]) red[tid] = o; }
            __syncthreads();
        }
        unsigned long long kmax = red[0];
        __syncthreads();
#pragma unroll
        for (int k = 0; k < 20; ++k) {
            int li = tid + k * 256;
            if (li < 5000 && cl[li] == kmax) cl[li] = 0ull;
        }
        if (tid == 0) topi[s] = ~((unsigned)(kmax & 0xffffffffull));
        __syncthreads();
    }
    if (tid < TOPK) {
        unsigned item = topi[tid];
        int h = 0;
        for (int j = 0; j < HIST; ++j)
            if ((unsigned)xl[j] == item + 1u) h = 1;   // history is 1-indexed
        hit[tid] = h;
    }
    __syncthreads();
    if (tid == 0) {
        int rsel = -1;
        for (int s = 0; s < TOPK; ++s) if (hit[s]) { rsel = s; break; }
        unsigned a = (rsel >= 0) ? topi[rsel] : topi[0];
        int fb = (rsel >= 0) ? 1 : -1;
        mask[a] = 0.0f;
        state[0] = (int)a;
        state[1] = fb;
        out[t] = (float)a;              // a_hats
        out[T_STEPS + t] = (float)fb;   // feedbacks
    }
}

// ---------------------------------------------------------------------------
extern "C" void kernel_launch(void* const* d_in, const int* in_sizes, int n_in,
                              void* d_out, int out_size, void* d_ws, size_t ws_size,
                              hipStream_t stream) {
    const int*   x      = (const int*)d_in[0];
    const float* embed  = (const float*)d_in[1];
    const float* W_ih   = (const float*)d_in[2];
    const float* W_hh   = (const float*)d_in[3];
    const float* b_ih   = (const float*)d_in[4];
    const float* b_hh   = (const float*)d_in[5];
    const float* W_out  = (const float*)d_in[6];
    const float* b_out  = (const float*)d_in[7];
    float* out = (float*)d_out;

    char* ws = (char*)d_ws;
    float*    mask  = (float*)(ws + OFF_MASK);
    float*    elog  = (float*)(ws + OFF_ELOG);
    float*    hbuf  = (float*)(ws + OFF_H);
    float*    cbuf  = (float*)(ws + OFF_C);
    unsigned* gmaxU = (unsigned*)(ws + OFF_SCAL);
    float*    gsum  = (float*)(ws + OFF_SCAL + 4);
    int*      state = (int*)(ws + OFF_STATE);
    unsigned long long* cand = (unsigned long long*)(ws + OFF_CAND);
    float*    part  = (float*)(ws + OFF_PART);

    k_init<<<(OUT_N + 255) / 256, 256, 0, stream>>>(mask, hbuf, cbuf, state);

    for (int t = 0; t < T_STEPS; ++t) {
        k_cell<<<1, 256, 0, stream>>>(embed, W_ih, W_hh, b_ih, b_hh,
                                      hbuf, cbuf, state, gmaxU, gsum, t);
        k_matvec<<<625, 256, 0, stream>>>(W_out, b_out, hbuf, elog, gmaxU);
        k_expsum<<<625, 256, 0, stream>>>(elog, gmaxU, part);
        k_sumreduce<<<1, 256, 0, stream>>>(part, gsum);
        float* out_masked = out + 2 * T_STEPS + (size_t)t * OUT_N;
        k_topk_local<<<100, 256, 0, stream>>>(elog, mask, gsum, out_masked, cand);
        k_final<<<1, 256, 0, stream>>>(cand, x, mask, state, out, t);
    }
}